// VectorQuantizer_3667902071452
// MI455X (gfx1250) — compile-verified
//
#include <hip/hip_runtime.h>

// ---------------------------------------------------------------------------
// VQ-VAE vector quantizer for MI455X (gfx1250, wave32, WMMA).
//   x:          [32, 64, 64, 64] f32 (NCHW)  -> flat_x [N=131072, D=64]
//   embeddings: [64, 512] f32               (D x K)
// Outputs (concatenated): quantized_ste [N*D], commitment_loss [1], idx [N]
//
// Kernel 1: distances via V_WMMA_F32_16X16X4_F32. A (= -2*x tile) lives in
// VGPRs; B (= embeddings) is staged in LDS as interleaved row-pairs so each
// 4-deep WMMA B fragment is a single conflict-free ds_load_b64. Each wave
// owns 32 rows (two M-tiles) -> 2 WMMAs per B load.
// ---------------------------------------------------------------------------

typedef float v2f __attribute__((ext_vector_type(2)));
typedef float v8f __attribute__((ext_vector_type(8)));

static constexpr int BATCH = 32;
static constexpr int D     = 64;
static constexpr int HGT   = 64;
static constexpr int WID   = 64;
static constexpr int HW    = HGT * WID;        // 4096
static constexpr int NROWS = BATCH * HW;       // 131072
static constexpr int K     = 512;              // codebook entries
static constexpr int CHW   = D * HW;           // 262144 (stride between batches)

// Interleaved pair layout: sE[p][k][2] with p = d/2 (32 pair-rows).
// Pair-row stride 1056 floats: 1056 % 64 == 32, so odd pair-rows (upper
// half-wave) land on the other 32 banks than even pair-rows -> conflict-free
// ds_load_b64 for the B operand.
static constexpr int PAIRSTRIDE       = 2 * K + 32;                  // 1056
static constexpr int LDS_E_FLOATS     = (D / 2) * PAIRSTRIDE;        // 33792
static constexpr int LDS_TOTAL_FLOATS = LDS_E_FLOATS + K;            // + ||e||^2

__global__ __launch_bounds__(256)
void vq_argmin_kernel(const float* __restrict__ x,
                      const float* __restrict__ emb,
                      int*   __restrict__ idx_out,
                      float* __restrict__ idx_out_f)
{
    extern __shared__ float smem[];
    float* sE    = smem;                 // [32][PAIRSTRIDE] interleaved pairs
    float* sNorm = smem + LDS_E_FLOATS;  // [K]

    const int tid = threadIdx.x;

    // Stage embeddings (128 KB) into the interleaved pair layout.
    for (int i = tid; i < D * K; i += 256) {
        const int d = i >> 9;            // / 512
        const int k = i & (K - 1);
        sE[(d >> 1) * PAIRSTRIDE + (k << 1) + (d & 1)] = emb[i];
    }
    __syncthreads();

    // Codebook squared norms (b64 reads of the pair layout).
    for (int k = tid; k < K; k += 256) {
        float s = 0.0f;
        #pragma unroll 8
        for (int p = 0; p < D / 2; ++p) {
            const v2f e = *(const v2f*)(sE + p * PAIRSTRIDE + (k << 1));
            s += e.x * e.x + e.y * e.y;
        }
        sNorm[k] = s;
    }
    __syncthreads();

    const int wave = tid >> 5;                 // 0..7
    const int lane = tid & 31;
    const int row  = lane & 15;
    const int dsel = (lane >> 4) << 1;         // 0 for lanes 0-15, 2 for 16-31
    const int m0   = (blockIdx.x * 8 + wave) * 32;   // 32 rows per wave

    // Two A operands (rows m0..+15 and m0+16..+31), prescaled by -2.
    // 16x4 f32 A layout: lanes 0-15 hold K=4kk+{0,1}; lanes 16-31 K=4kk+{2,3}.
    const int mA = m0 + row;
    const int mB = m0 + 16 + row;
    const long xbaseA = (long)(mA >> 12) * CHW + (mA & (HW - 1));
    const long xbaseB = (long)(mB >> 12) * CHW + (mB & (HW - 1));

    v2f a0[16], a1[16];
    #pragma unroll
    for (int kk = 0; kk < 16; ++kk) {
        const int d0 = kk * 4 + dsel;
        v2f v;
        v.x = -2.0f * x[xbaseA + (long)d0 * HW];
        v.y = -2.0f * x[xbaseA + (long)(d0 + 1) * HW];
        a0[kk] = v;
        v.x = -2.0f * x[xbaseB + (long)d0 * HW];
        v.y = -2.0f * x[xbaseB + (long)(d0 + 1) * HW];
        a1[kk] = v;
    }

    float bestv0[8], bestv1[8];
    int   besti0[8], besti1[8];
    #pragma unroll
    for (int r = 0; r < 8; ++r) {
        bestv0[r] = 3.4e38f; besti0[r] = 0;
        bestv1[r] = 3.4e38f; besti1[r] = 0;
    }

    // Per-lane B base: pair-row (dsel/2), column (row). Split into two bases
    // so every in-loop DS offset is a 16-bit immediate (max 8448*7 = 59136 B).
    const float* bBase = sE + (dsel >> 1) * PAIRSTRIDE + (row << 1);

    for (int n0 = 0; n0 < K; n0 += 16) {
        const float* bcLo = bBase + (n0 << 1);                  // pairs 0..14
        const float* bcHi = bcLo + 8 * 2 * PAIRSTRIDE;          // pairs 16..30
        v8f acc0 = {};
        v8f acc1 = {};
        #pragma unroll
        for (int kk = 0; kk < 8; ++kk) {
            const v2f b = *(const v2f*)(bcLo + kk * 2 * PAIRSTRIDE);
            acc0 = __builtin_amdgcn_wmma_f32_16x16x4_f32(
                false, a0[kk], false, b, (short)0, acc0, false, false);
            acc1 = __builtin_amdgcn_wmma_f32_16x16x4_f32(
                false, a1[kk], false, b, (short)0, acc1, false, false);
        }
        #pragma unroll
        for (int kk = 0; kk < 8; ++kk) {
            const v2f b = *(const v2f*)(bcHi + kk * 2 * PAIRSTRIDE);
            acc0 = __builtin_amdgcn_wmma_f32_16x16x4_f32(
                false, a0[kk + 8], false, b, (short)0, acc0, false, false);
            acc1 = __builtin_amdgcn_wmma_f32_16x16x4_f32(
                false, a1[kk + 8], false, b, (short)0, acc1, false, false);
        }

        // C layout: VGPR r, lane<16 -> (M=r, N=n0+lane); lane>=16 -> (M=r+8).
        const float en = sNorm[n0 + row];
        const int   n  = n0 + row;
        #pragma unroll
        for (int r = 0; r < 8; ++r) {
            const float s0 = acc0[r] + en;     // -2 x.e + ||e||^2
            if (s0 < bestv0[r]) { bestv0[r] = s0; besti0[r] = n; }
            const float s1 = acc1[r] + en;
            if (s1 < bestv1[r]) { bestv1[r] = s1; besti1[r] = n; }
        }
    }

    // Reduce (min,argmin) across the 16 lanes of each half; lowest-index ties
    // match jnp.argmin. XOR masks < 16 never cross the halves.
    #pragma unroll
    for (int off = 8; off >= 1; off >>= 1) {
        #pragma unroll
        for (int r = 0; r < 8; ++r) {
            float ov = __shfl_xor(bestv0[r], off, 32);
            int   oi = __shfl_xor(besti0[r], off, 32);
            if (ov < bestv0[r] || (ov == bestv0[r] && oi < besti0[r])) {
                bestv0[r] = ov; besti0[r] = oi;
            }
            ov = __shfl_xor(bestv1[r], off, 32);
            oi = __shfl_xor(besti1[r], off, 32);
            if (ov < bestv1[r] || (ov == bestv1[r] && oi < besti1[r])) {
                bestv1[r] = ov; besti1[r] = oi;
            }
        }
    }

    if (row == 0) {
        const int half = (lane >> 4) * 8;      // lane 0 -> rows r, lane 16 -> r+8
        #pragma unroll
        for (int r = 0; r < 8; ++r) {
            const int mm0 = m0 + half + r;          // tile 0
            idx_out[mm0]   = besti0[r];
            idx_out_f[mm0] = (float)besti0[r];
            const int mm1 = m0 + 16 + half + r;     // tile 1
            idx_out[mm1]   = besti1[r];
            idx_out_f[mm1] = (float)besti1[r];
        }
    }
}

// ---------------------------------------------------------------------------
// Kernel 2: gather codebook rows into NCHW output + per-block loss partials.
// ---------------------------------------------------------------------------
__global__ __launch_bounds__(256)
void vq_gather_kernel(const float* __restrict__ x,
                      const float* __restrict__ emb,
                      const int*  __restrict__ idx,
                      float* __restrict__ quant_out,
                      float* __restrict__ partial)
{
    __shared__ float sred[256];
    const int n    = blockIdx.x * 256 + threadIdx.x;
    const int code = idx[n];
    const long xbase = (long)(n >> 12) * CHW + (n & (HW - 1));

    float s = 0.0f;
    #pragma unroll 4
    for (int d = 0; d < D; ++d) {
        const float q  = emb[d * K + code];
        const long  a  = xbase + (long)d * HW;
        const float xv = x[a];
        quant_out[a] = q;                      // STE output == quantized
        const float diff = xv - q;
        s += diff * diff;
    }

    sred[threadIdx.x] = s;
    __syncthreads();
    #pragma unroll
    for (int st = 128; st >= 1; st >>= 1) {
        if (threadIdx.x < st) sred[threadIdx.x] += sred[threadIdx.x + st];
        __syncthreads();
    }
    if (threadIdx.x == 0) partial[blockIdx.x] = sred[0];
}

// ---------------------------------------------------------------------------
// Kernel 3: deterministic final reduction of 512 partials -> mean loss.
// ---------------------------------------------------------------------------
__global__ __launch_bounds__(256)
void vq_loss_kernel(const float* __restrict__ partial, float* __restrict__ loss)
{
    __shared__ float sred[256];
    sred[threadIdx.x] = partial[threadIdx.x] + partial[threadIdx.x + 256];
    __syncthreads();
    #pragma unroll
    for (int st = 128; st >= 1; st >>= 1) {
        if (threadIdx.x < st) sred[threadIdx.x] += sred[threadIdx.x + st];
        __syncthreads();
    }
    if (threadIdx.x == 0) *loss = sred[0] / (float)(NROWS * D);
}

// ---------------------------------------------------------------------------
extern "C" void kernel_launch(void* const* d_in, const int* in_sizes, int n_in,
                              void* d_out, int out_size, void* d_ws, size_t ws_size,
                              hipStream_t stream)
{
    const float* x   = (const float*)d_in[0];
    const float* emb = (const float*)d_in[1];

    float* out   = (float*)d_out;
    float* quant = out;                              // [N*D]  = 8388608
    float* loss  = out + (size_t)NROWS * D;          // [1]
    float* idx_f = loss + 1;                         // [N]    = 131072

    int*   idx_i   = (int*)d_ws;                                   // 512 KB
    float* partial = (float*)((char*)d_ws + (size_t)NROWS * sizeof(int)); // 2 KB

    const size_t smem1 = (size_t)LDS_TOTAL_FLOATS * sizeof(float); // ~134 KB

    vq_argmin_kernel<<<NROWS / 256, 256, smem1, stream>>>(x, emb, idx_i, idx_f);
    vq_gather_kernel<<<NROWS / 256, 256, 0, stream>>>(x, emb, idx_i, quant, partial);
    vq_loss_kernel<<<1, 256, 0, stream>>>(partial, loss);
}